// ConLoss_15899968930060
// MI455X (gfx1250) — compile-verified
//
#include <hip/hip_runtime.h>
#include <math.h>

// ---------------------------------------------------------------------------
// Supervised-contrastive prototype loss, collapsed form.
//   S[c,j] = protos[c] . contrast[j] / T          (100x8192, K=512, f32 WMMA)
//   per class: m_c, E_c = sum exp(S-m), T_c = masked sum, match_c = 2*cnt_c
//   per anchor i (class c): (T_c - g_i)/(match_c-1) - m_c - log(E_c - exp(g_i-m_c))
//   loss = -mean over 8192 anchors
// ---------------------------------------------------------------------------

typedef float v2f __attribute__((ext_vector_type(2)));
typedef float v8f __attribute__((ext_vector_type(8)));

#define TEMP_INV (1.0f / 0.07f)
#define B_N   4096      // batch
#define D_K   512       // feature dim
#define NCLS  100
#define COLS  8192      // V*B
#define CHUNK 1024      // columns per block
#define NCHUNK (COLS / CHUNK)   // 8
#define NCGRP  7                // ceil(100/16) class-row tiles

__device__ __forceinline__ float wred_max(float v) {
  #pragma unroll
  for (int o = 16; o > 0; o >>= 1) v = fmaxf(v, __shfl_xor(v, o, 32));
  return v;
}
__device__ __forceinline__ float wred_sum(float v) {
  #pragma unroll
  for (int o = 16; o > 0; o >>= 1) v += __shfl_xor(v, o, 32);
  return v;
}

__global__ void k_init(float* acc) { *acc = 0.0f; }

// grid (7, 8) x 256 threads. Block (bx,by): class rows [16bx,16bx+16),
// columns [1024by, 1024by+1024). 8 waves, each owns 8 16x16 WMMA tiles.
__global__ __launch_bounds__(256) void k_gemm_stats(
    const float* __restrict__ feat,      // [4096,2,512]
    const int*   __restrict__ labels,    // [4096]
    const float* __restrict__ protos,    // [100,512]
    float*       __restrict__ g,         // [8192] gathered S[labels[j%B], j]
    float4*      __restrict__ pstats)    // [112][8] (m, e, T, cnt)
{
  __shared__ float sS[16 * CHUNK];       // 64 KB chunk of S, scaled by 1/T
  const int wave = threadIdx.x >> 5;
  const int lane = threadIdx.x & 31;
  const int c0 = blockIdx.x * 16;
  const int colbase = blockIdx.y * CHUNK;
  const int n    = lane & 15;            // M for A-frag, N for B-frag
  const int half = lane >> 4;            // K-half select (ISA 16x4 f32 layout)

  // A: 16 proto rows (clamp padded rows 100..111 to row 99; never consumed).
  int rowA = c0 + n; if (rowA > NCLS - 1) rowA = NCLS - 1;
  const float* aptr = protos + (size_t)rowA * D_K + half * 2;

  for (int t = wave; t < CHUNK / 16; t += 8) {
    const int j = colbase + t * 16 + n;  // global contrast column
    // contrast[j] = features[j%B, j/B, :]
    const float* bptr = feat + (size_t)(j & (B_N - 1)) * (2 * D_K)
                             + (size_t)(j >> 12) * D_K + half * 2;
    v8f acc = {};
    #pragma unroll 4
    for (int k = 0; k < D_K; k += 4) {
      v2f a = *(const v2f*)(aptr + k);   // A[m, k0+2*half .. +1]
      v2f b = *(const v2f*)(bptr + k);   // B[k0+2*half .. +1, n]
      acc = __builtin_amdgcn_wmma_f32_16x16x4_f32(
          false, a, false, b, (short)0, acc, false, false);
    }
    const int coll = t * 16 + n;
    #pragma unroll
    for (int v = 0; v < 8; ++v)          // C layout: VGPR v -> row v + 8*half
      sS[(v + half * 8) * CHUNK + coll] = acc[v] * TEMP_INV;
  }
  __syncthreads();

  // Per-wave stats: wave w owns rows 2w, 2w+1 of the chunk.
  #pragma unroll
  for (int rr = 0; rr < 2; ++rr) {
    const int r = 2 * wave + rr;
    const int cls = c0 + r;
    float lmax = -3.402823466e38f, lT = 0.0f, lcnt = 0.0f;
    for (int nl = lane; nl < CHUNK; nl += 32) {
      const float s = sS[r * CHUNK + nl];
      lmax = fmaxf(lmax, s);
      const int j = colbase + nl;
      if (labels[j & (B_N - 1)] == cls) {  // positive-mask column
        lT += s; lcnt += 1.0f;
        g[j] = s;                          // unique writer: cls == labels[j%B]
      }
    }
    const float m = wred_max(lmax);
    float lsum = 0.0f;
    for (int nl = lane; nl < CHUNK; nl += 32)
      lsum += expf(sS[r * CHUNK + nl] - m);
    const float e   = wred_sum(lsum);
    const float T   = wred_sum(lT);
    const float cnt = wred_sum(lcnt);
    if (lane == 0)
      pstats[cls * NCHUNK + blockIdx.y] = make_float4(m, e, T, cnt);
  }
}

// Merge 8 chunk partials per class (online-softmax merge). 1 block x 128.
__global__ void k_merge(const float4* __restrict__ pstats,
                        float4* __restrict__ stats)
{
  const int c = threadIdx.x;
  if (c >= NCGRP * 16) return;
  float M = -3.402823466e38f;
  #pragma unroll
  for (int k = 0; k < NCHUNK; ++k) M = fmaxf(M, pstats[c * NCHUNK + k].x);
  float E = 0.0f, T = 0.0f, cnt = 0.0f;
  #pragma unroll
  for (int k = 0; k < NCHUNK; ++k) {
    const float4 p = pstats[c * NCHUNK + k];
    E += p.y * expf(p.x - M);
    T += p.z; cnt += p.w;
  }
  stats[c] = make_float4(M, E, T, cnt);
}

// Per-anchor terms + block reduction + atomic accumulate. 32 blocks x 256.
__global__ __launch_bounds__(256) void k_loss(
    const int* __restrict__ labels, const float* __restrict__ g,
    const float4* __restrict__ stats, float* __restrict__ acc)
{
  const int i = blockIdx.x * 256 + threadIdx.x;
  float term = 0.0f;
  if (i < COLS) {
    const int c = labels[i & (B_N - 1)];
    const float4 st = stats[c];                 // (m, E, T, match)
    const float gi = g[i];
    const float denom = st.y - expf(gi - st.x); // exp-sum minus self-contrast
    term = (st.z - gi) / (st.w - 1.0f) - st.x - logf(denom);
  }
  const int wave = threadIdx.x >> 5, lane = threadIdx.x & 31;
  float s = wred_sum(term);
  __shared__ float wsum[8];
  if (lane == 0) wsum[wave] = s;
  __syncthreads();
  if (threadIdx.x == 0) {
    float b = 0.0f;
    #pragma unroll
    for (int w = 0; w < 8; ++w) b += wsum[w];
    atomicAdd(acc, b);
  }
}

__global__ void k_finalize(const float* acc, float* out) {
  out[0] = -acc[0] * (1.0f / (float)COLS);
}

extern "C" void kernel_launch(void* const* d_in, const int* in_sizes, int n_in,
                              void* d_out, int out_size, void* d_ws, size_t ws_size,
                              hipStream_t stream) {
  const float* feat   = (const float*)d_in[0];   // [4096,2,512] f32
  const int*   labels = (const int*)  d_in[1];   // [4096] int
  const float* protos = (const float*)d_in[2];   // [100,512] f32
  float* out = (float*)d_out;

  char* ws = (char*)d_ws;
  float*  g      = (float*) (ws);                        // 8192 f32 = 32768 B
  float4* pstats = (float4*)(ws + 32768);                // 112*8*16 = 14336 B
  float4* stats  = (float4*)(ws + 32768 + 14336);        // 112*16   =  1792 B
  float*  acc    = (float*) (ws + 32768 + 14336 + 1792); // 4 B

  k_init<<<1, 1, 0, stream>>>(acc);
  k_gemm_stats<<<dim3(NCGRP, NCHUNK), 256, 0, stream>>>(feat, labels, protos,
                                                        g, pstats);
  k_merge<<<1, 128, 0, stream>>>(pstats, stats);
  k_loss<<<COLS / 256, 256, 0, stream>>>(labels, g, stats, acc);
  k_finalize<<<1, 1, 0, stream>>>(acc, out);
}